// kernel_generated_0_62904091017438
// MI455X (gfx1250) — compile-verified
//
#include <hip/hip_runtime.h>

typedef float v2f __attribute__((ext_vector_type(2)));
typedef float v8f __attribute__((ext_vector_type(8)));

#define CH 128
#define HH 56
#define WW 56
#define PLANE (HH * WW)            // 3136
#define IMG (CH * PLANE)           // 401408

// ---------------------------------------------------------------------------
// Stage 1: t4[l,a,m,o,j] = sum_{c,i} w1[c,i,j] * T(l, 2c+a, m+i-1, o)
// via V_WMMA_F32_16X16X4_F32:
//   M = 16 'o' positions, N = (a,j) packed in cols 0..7, K = channel chunks.
// Block = 4 waves, one per o-tile {0,16,32,40}; blockIdx.x = l*56 + m.
// Output layout: t4ws[((l*56+m)*56 + o)*8 + (a*4+j)]  (12.8 MB in d_ws)
// ---------------------------------------------------------------------------
__global__ __launch_bounds__(128) void stage1_wmma(
    const float* __restrict__ x, const float* __restrict__ w1,
    float* __restrict__ t4)
{
  __shared__ float w1s[768];                       // w1: (64,3,4)
  const int tid = threadIdx.y * 32 + threadIdx.x;
  for (int i = tid; i < 768; i += 128) w1s[i] = w1[i];
  __syncthreads();

  const int m  = blockIdx.x % 56;
  const int l  = blockIdx.x / 56;
  const int o0 = (threadIdx.y < 3) ? (int)threadIdx.y * 16 : 40; // last tile overlaps (same values)

  const int lane = threadIdx.x;
  const int hi   = lane >> 4;      // K half: lanes 0-15 -> K{0,1}, 16-31 -> K{2,3}
  const int lo   = lane & 15;      // A: M row;  B/D: N column
  const int o    = o0 + lo;
  const int aa   = (lo >> 2) & 1;  // N = a*4 + j for N<8
  const int jj   = lo & 3;
  const float bm0 = (lo < 8 && aa == 0) ? 1.0f : 0.0f;  // K even -> parity-0 channels
  const float bm1 = (lo < 8 && aa == 1) ? 1.0f : 0.0f;  // K odd  -> parity-1 channels

  v8f acc = {0.f, 0.f, 0.f, 0.f, 0.f, 0.f, 0.f, 0.f};

  const float* xl = x + (size_t)l * IMG;

  #pragma unroll
  for (int i = 0; i < 3; ++i) {
    const int h = m + i - 1;
    if (h < 0 || h >= HH) continue;            // uniform branch: EXEC stays all-ones
    const int r = (h + 55) % 56;               // apply roll(+1): row h reads x row (h-1)%56
    const float* xrow = xl + r * WW + o;
    const float* wcol = &w1s[hi * 12 + i * 4 + jj];
    #pragma unroll 8
    for (int s = 0; s < 32; ++s) {             // K = 4 channels per WMMA, 128 channels total
      const int ch0 = 4 * s + 2 * hi;
      v2f af;                                  // A[M=lo, K=2*hi+v] = x[ch0+v, r, o]
      af.x = xrow[ch0 * PLANE];
      af.y = xrow[(ch0 + 1) * PLANE];
      const float wv = wcol[s * 24];           // w1s[(2s+hi)*12 + i*4 + jj]
      v2f bf;                                  // B[K, N=(a,j)] nonzero iff (K parity == a)
      bf.x = wv * bm0;
      bf.y = wv * bm1;
      acc = __builtin_amdgcn_wmma_f32_16x16x4_f32(
          false, af, false, bf, (short)0, acc, false, false);
    }
  }

  // D layout: lane holds column N=lo; VGPR v holds row M = v + 8*hi
  if (lo < 8) {
    float* trow = t4 + ((size_t)(l * 56 + m) * 56) * 8;
    #pragma unroll
    for (int v = 0; v < 8; ++v) {
      const int orow = o0 + v + 8 * hi;
      trow[orow * 8 + lo] = acc[v];
    }
  }
}

// ---------------------------------------------------------------------------
// Stage 2: out[l, j2*4+j, m, n] = sum_{a,k} w2[a,k,j2] * t4[l,a,m,n+k-1,j]
// Store-bandwidth bound (205 MB out): lanes vary n -> 128B coalesced stores.
// Block (64,4): tx=n (56 used), ty=j; blockIdx.x = l*56 + m.
// ---------------------------------------------------------------------------
__global__ __launch_bounds__(256) void stage2(
    const float* __restrict__ t4, const float* __restrict__ w2,
    float* __restrict__ out)
{
  __shared__ float w2s[192];                   // w2: (2,3,32)
  const int tid = threadIdx.y * 64 + threadIdx.x;
  if (tid < 192) w2s[tid] = w2[tid];
  __syncthreads();

  const int m = blockIdx.x % 56;
  const int l = blockIdx.x / 56;
  const int n = threadIdx.x;
  const int j = threadIdx.y;
  if (n >= 56) return;

  const float* trow = t4 + ((size_t)(l * 56 + m) * 56) * 8;
  float v0[3], v1[3];
  #pragma unroll
  for (int k = 0; k < 3; ++k) {
    const int nn = n + k - 1;
    if (nn >= 0 && nn < 56) {
      v0[k] = trow[nn * 8 + j];                // a = 0
      v1[k] = trow[nn * 8 + 4 + j];            // a = 1
    } else {
      v0[k] = 0.f; v1[k] = 0.f;
    }
  }

  float* op = out + (size_t)l * IMG + (size_t)j * PLANE + m * WW + n;
  #pragma unroll 4
  for (int j2 = 0; j2 < 32; ++j2) {
    float accv = v0[0] * w2s[j2]       + v0[1] * w2s[32 + j2] + v0[2] * w2s[64 + j2]
               + v1[0] * w2s[96 + j2]  + v1[1] * w2s[128 + j2] + v1[2] * w2s[160 + j2];
    op[(size_t)(j2 * 4) * PLANE] = accv;       // channel C = j2*4 + j
  }
}

// ---------------------------------------------------------------------------
extern "C" void kernel_launch(void* const* d_in, const int* in_sizes, int n_in,
                              void* d_out, int out_size, void* d_ws, size_t ws_size,
                              hipStream_t stream)
{
  const float* x  = (const float*)d_in[0];   // (128,128,56,56)
  const float* w1 = (const float*)d_in[1];   // (64,3,4)
  const float* w2 = (const float*)d_in[2];   // (2,3,32)
  float* out = (float*)d_out;                // (128,128,56,56)
  float* t4  = (float*)d_ws;                 // needs 128*56*56*8*4 B = 12.85 MB

  dim3 b1(32, 4);
  dim3 g1(128 * 56);
  stage1_wmma<<<g1, b1, 0, stream>>>(x, w1, t4);

  dim3 b2(64, 4);
  dim3 g2(128 * 56);
  stage2<<<g2, b2, 0, stream>>>(t4, w2, out);
}